// MSTransPoseNet_12455405158402
// MI455X (gfx1250) — compile-verified
//
#include <hip/hip_runtime.h>
#include <hip/hip_bf16.h>
#include <math.h>

// Problem constants (from reference): B=2048, S=16, D=512, CH=1024
#define B_  2048
#define S_  16
#define D_  512
#define CH_ 1024

typedef __attribute__((ext_vector_type(2))) float v2f;
typedef __attribute__((ext_vector_type(8))) float v8f;

// LDS row strides (padded to dodge bank conflicts, keep 8B alignment)
#define XSTRIDE 520    // >= 512, even, 4-float aligned
#define HSTRIDE 1032   // >= 1024, even, 8-word bank shift between rows

// ---------------------------------------------------------------------------
// Kernel 1: scene router.  One block (16 waves = 512 thr) per sample.
//   wave s computes logit[s] = dot(cat[b,s,:], w_scene) + b_scene
//   wave 0: log_softmax -> d_out, argmax -> per-scene index lists (workspace)
// ---------------------------------------------------------------------------
__global__ void router_kernel(const float* __restrict__ descs_t,
                              const float* __restrict__ descs_rot,
                              const float* __restrict__ w_scene,
                              const float* __restrict__ b_scene,
                              int* __restrict__ counts,
                              int* __restrict__ lists,
                              float* __restrict__ out)
{
    const int b    = blockIdx.x;
    const int tid  = threadIdx.x;
    const int wave = tid >> 5;      // scene index, 0..15
    const int lane = tid & 31;

    __shared__ float s_logit[S_];

    const float* tp = descs_t   + (size_t)b * S_ * D_ + (size_t)wave * D_;
    const float* rp = descs_rot + (size_t)b * S_ * D_ + (size_t)wave * D_;

    float acc = 0.f;
    #pragma unroll 4
    for (int d = lane; d < D_; d += 32) {
        acc += tp[d] * w_scene[d];
        acc += rp[d] * w_scene[D_ + d];
    }
    #pragma unroll
    for (int off = 16; off > 0; off >>= 1) acc += __shfl_down(acc, off, 32);
    if (lane == 0) s_logit[wave] = acc + b_scene[0];
    __syncthreads();

    if (wave == 0) {
        const float NEG = -__builtin_inff();
        float lg = (lane < S_) ? s_logit[lane] : NEG;

        float mx = lg;
        #pragma unroll
        for (int off = 16; off > 0; off >>= 1)
            mx = fmaxf(mx, __shfl_down(mx, off, 32));
        mx = __shfl(mx, 0, 32);

        float ex = (lane < S_) ? expf(lg - mx) : 0.f;
        float se = ex;
        #pragma unroll
        for (int off = 16; off > 0; off >>= 1) se += __shfl_down(se, off, 32);
        se = __shfl(se, 0, 32);
        const float lse = logf(se);

        if (lane < S_)
            out[(size_t)B_ * 7 + (size_t)b * S_ + lane] = lg - mx - lse;

        // argmax with first-index tie-break (matches jnp.argmax)
        int idx = (lane < S_ && lg == mx) ? lane : S_;
        #pragma unroll
        for (int off = 16; off > 0; off >>= 1)
            idx = min(idx, __shfl_down(idx, off, 32));
        if (lane == 0) {
            int pos = atomicAdd(&counts[idx], 1);
            lists[idx * B_ + pos] = b;
        }
    }
}

// ---------------------------------------------------------------------------
// Kernel 2: grouped expert heads.
//   grid = (16 scenes, 128 row-tiles); block = 256 thr (8 waves).
//   Per block: gather <=16 rows of scene s into LDS, GEMM [16,512]x[512,1024]
//   with V_WMMA_F32_16X16X4_F32, fused bias+GELU, then [16,1024]x[1024,nout].
// ---------------------------------------------------------------------------
__global__ void expert_kernel(const float* __restrict__ descs_t,
                              const float* __restrict__ descs_rot,
                              const float* __restrict__ Wh_t,
                              const float* __restrict__ bh_t,
                              const float* __restrict__ Wo_t,
                              const float* __restrict__ bo_t,
                              const float* __restrict__ Wh_r,
                              const float* __restrict__ bh_r,
                              const float* __restrict__ Wo_r,
                              const float* __restrict__ bo_r,
                              const int* __restrict__ counts,
                              const int* __restrict__ lists,
                              float* __restrict__ out)
{
    extern __shared__ float smem[];
    float* Xs = smem;                     // [16][XSTRIDE]  gathered descriptors
    float* Hs = smem + 16 * XSTRIDE;      // [16][HSTRIDE]  hidden activations
    __shared__ int s_idx[16];

    const int s     = blockIdx.x;
    const int tile  = blockIdx.y;
    const int count = counts[s];
    const int row0  = tile * 16;
    if (row0 >= count) return;            // uniform per block
    const int nrows = min(16, count - row0);

    const int tid  = threadIdx.x;
    const int lane = tid & 31;
    const int wave = tid >> 5;            // 0..7
    const int col  = lane & 15;           // N col in tile / M row for A frag
    const int hi   = lane >> 4;           // 0: K{0,1}, 1: K{2,3}

    if (tid < 16) {
        int r = (tid < nrows) ? tid : 0;  // pad with a valid row
        s_idx[tid] = lists[s * B_ + row0 + r];
    }
    __syncthreads();

    const v8f vzero = {0.f, 0.f, 0.f, 0.f, 0.f, 0.f, 0.f, 0.f};

    for (int head = 0; head < 2; ++head) {
        const float* src = head ? descs_rot : descs_t;
        const float* Wh  = head ? Wh_r : Wh_t;
        const float* bh  = head ? bh_r : bh_t;
        const float* Wo  = head ? Wo_r : Wo_t;
        const float* bo  = head ? bo_r : bo_t;
        const int   nout = head ? 4 : 3;

        // ---- stage gathered X rows into LDS (float4, coalesced) ----
        for (int i = tid; i < 16 * (D_ / 4); i += 256) {
            int m  = i >> 7;              // / 128
            int d4 = i & 127;
            const float4* p =
                (const float4*)(src + (size_t)s_idx[m] * S_ * D_ + (size_t)s * D_);
            ((float4*)(Xs + m * XSTRIDE))[d4] = p[d4];
        }
        __syncthreads();

        // ---- hidden GEMM via f32 WMMA; wave owns N-panel [wave*128, +128) ----
        const float* WhS = Wh + (size_t)s * D_ * CH_;
        v8f acc[8];
        #pragma unroll
        for (int t = 0; t < 8; ++t) acc[t] = vzero;

        const int npan = wave * 128 + col;
        for (int k = 0; k < D_; k += 4) {
            const int ka = k + 2 * hi;
            // A fragment: X[m=col][ka], X[m=col][ka+1]
            v2f a = *(const v2f*)(Xs + col * XSTRIDE + ka);
            const float* brow = WhS + (size_t)ka * CH_ + npan;
            #pragma unroll
            for (int t = 0; t < 8; ++t) {
                v2f bfrag;
                bfrag.x = brow[t * 16];        // row ka,   col npan+16t
                bfrag.y = brow[t * 16 + CH_];  // row ka+1, col npan+16t
                acc[t] = __builtin_amdgcn_wmma_f32_16x16x4_f32(
                    false, a, false, bfrag, (short)0, acc[t], false, false);
            }
        }

        // ---- epilogue: bias + exact GELU, store H to LDS ----
        #pragma unroll
        for (int t = 0; t < 8; ++t) {
            const int   n    = wave * 128 + t * 16 + col;
            const float bias = bh[s * CH_ + n];
            #pragma unroll
            for (int e = 0; e < 8; ++e) {
                const int m = e + 8 * hi;
                float x = acc[t][e] + bias;
                float g = 0.5f * x * (1.0f + erff(x * 0.70710678118654752f));
                Hs[m * HSTRIDE + n] = g;
            }
        }
        __syncthreads();

        // ---- output GEMM [16,1024]x[1024,nout]: wave-reduced VALU dots ----
        const float* WoS = Wo + (size_t)s * CH_ * nout;
        for (int mm = wave; mm < 16; mm += 8) {
            if (mm < nrows) {
                float sums[4] = {0.f, 0.f, 0.f, 0.f};
                for (int h = lane; h < CH_; h += 32) {
                    float hv = Hs[mm * HSTRIDE + h];
                    #pragma unroll
                    for (int j = 0; j < 4; ++j)
                        if (j < nout) sums[j] += hv * WoS[h * nout + j];
                }
                #pragma unroll
                for (int j = 0; j < 4; ++j) {
                    float v = sums[j];
                    #pragma unroll
                    for (int off = 16; off > 0; off >>= 1)
                        v += __shfl_down(v, off, 32);
                    sums[j] = v;
                }
                if (lane == 0) {
                    const int bidx = s_idx[mm];
                    for (int j = 0; j < nout; ++j)
                        out[(size_t)bidx * 7 + (head ? 3 + j : j)] =
                            sums[j] + bo[s * nout + j];
                }
            }
        }
        __syncthreads();   // before next head reuses Xs/Hs
    }
}

// ---------------------------------------------------------------------------
extern "C" void kernel_launch(void* const* d_in, const int* in_sizes, int n_in,
                              void* d_out, int out_size, void* d_ws, size_t ws_size,
                              hipStream_t stream)
{
    (void)in_sizes; (void)n_in; (void)out_size; (void)ws_size;

    const float* descs_t   = (const float*)d_in[0];
    const float* descs_rot = (const float*)d_in[1];
    const float* w_scene   = (const float*)d_in[2];
    const float* b_scene   = (const float*)d_in[3];
    const float* Wh_t      = (const float*)d_in[4];
    const float* bh_t      = (const float*)d_in[5];
    const float* Wo_t      = (const float*)d_in[6];
    const float* bo_t      = (const float*)d_in[7];
    const float* Wh_r      = (const float*)d_in[8];
    const float* bh_r      = (const float*)d_in[9];
    const float* Wo_r      = (const float*)d_in[10];
    const float* bo_r      = (const float*)d_in[11];
    float* out = (float*)d_out;

    int* counts = (int*)d_ws;          // [16]
    int* lists  = (int*)d_ws + 16;     // [16][2048]

    // zero the per-scene counters (graph-capture safe)
    hipMemsetAsync(counts, 0, S_ * sizeof(int), stream);

    router_kernel<<<B_, 512, 0, stream>>>(descs_t, descs_rot, w_scene, b_scene,
                                          counts, lists, out);

    const size_t smem = (size_t)(16 * XSTRIDE + 16 * HSTRIDE) * sizeof(float);
    expert_kernel<<<dim3(S_, B_ / 16), 256, smem, stream>>>(
        descs_t, descs_rot,
        Wh_t, bh_t, Wo_t, bo_t,
        Wh_r, bh_r, Wo_r, bo_r,
        counts, lists, out);
}